// CameraOnlyBEV_47107201302835
// MI455X (gfx1250) — compile-verified
//
#include <hip/hip_runtime.h>
#include <hip/hip_bf16.h>
#include <math.h>

typedef float v2f __attribute__((ext_vector_type(2)));
typedef float v8f __attribute__((ext_vector_type(8)));

#define FH_   64
#define FW_   176
#define HW    11264      // 64*176
#define CIN   256
#define CMID  128
#define DBINS 64
#define NB    4
#define NPIX  45056      // NB*HW
#define NGROUPS 2816     // NPIX/16
#define NPTS  150000

// ---------------------------------------------------------------------------
// K0: zero the BEV occupancy region of d_out
// ---------------------------------------------------------------------------
__global__ void zero_bev_kernel(float* __restrict__ bev) {
    int i = blockIdx.x * blockDim.x + threadIdx.x;
    if (i < NB * 200 * 200) bev[i] = 0.0f;
}

// ---------------------------------------------------------------------------
// K1: Conv2d(256->128, 3x3, pad=1) + bias, implicit GEMM via WMMA f32 16x16x4
//     GEMM: C[m,n] = sum_k A[m,k]*B[k,n],  k = r*256 + ci  (r = 3x3 tap)
//     One wave computes a 16(M) x 64(N) tile (4 accumulators).
//     Boundary handling is branch-free: base offsets are clamped to a valid
//     address once per (r,t) and loaded values are multiplied by a {0,1}
//     mask, so the inner K-loop has no EXEC manipulation (WMMA needs
//     EXEC == all ones anyway).
// ---------------------------------------------------------------------------
__global__ __launch_bounds__(32) void conv1_wmma_kernel(
    const float* __restrict__ X,   // [4,256,64,176]
    const float* __restrict__ W1,  // [128,256,3,3] -> A[m][ci*9+r]
    const float* __restrict__ b1,  // [128]
    float* __restrict__ Hbuf)      // [4,128,64,176]
{
    const int lane = threadIdx.x;
    const int ln = lane & 15;       // M (A) / N (B) index within tile
    const int hi = lane >> 4;       // half-wave select
    const int kk = hi * 2;          // local K offset of VGPR v=0 in the fragment
    const int m0 = blockIdx.y * 16; // output-channel tile base
    const int m  = m0 + ln;
    const int n0 = blockIdx.x * 64; // pixel-chunk base (never crosses a batch)

    int bI[4], yI[4], xI[4];
#pragma unroll
    for (int t = 0; t < 4; ++t) {
        int nt = n0 + t * 16;
        bI[t] = nt / HW;
        int rem = nt % HW;
        yI[t] = rem / FW_;
        xI[t] = rem % FW_;          // FW_%16==0 -> group sits in one row
    }

    v8f acc0 = {}, acc1 = {}, acc2 = {}, acc3 = {};

    for (int r = 0; r < 9; ++r) {
        const int dy = r / 3 - 1;
        const int dx = r % 3 - 1;
        // A element: W1[m*2304 + ci*9 + r], ci = ci0 + kk + v
        const float* Ap = W1 + (m * 2304 + kk * 9 + r);

        long  boff[4];
        float mk[4];
#pragma unroll
        for (int t = 0; t < 4; ++t) {
            int yy = yI[t] + dy;
            int xx = xI[t] + dx + ln;
            bool ok = ((unsigned)yy < (unsigned)FH_) && ((unsigned)xx < (unsigned)FW_);
            long off = ((long)(bI[t] * CIN + kk)) * HW + (long)yy * FW_ + xx;
            mk[t]   = ok ? 1.0f : 0.0f;
            boff[t] = ok ? off : 0;            // clamped: strided walk stays in X
            __builtin_prefetch(&X[boff[t]], 0, 3);
        }

        for (int ci = 0; ci < CIN; ci += 4) {
            v2f a;
            a.x = Ap[0];            // ci = ci0+kk
            a.y = Ap[9];            // ci = ci0+kk+1
            Ap += 36;               // advance ci by 4

            v2f b0, b1v, b2, b3;
            b0.x  = X[boff[0]]      * mk[0];
            b0.y  = X[boff[0] + HW] * mk[0];  boff[0] += 4L * HW;
            b1v.x = X[boff[1]]      * mk[1];
            b1v.y = X[boff[1] + HW] * mk[1];  boff[1] += 4L * HW;
            b2.x  = X[boff[2]]      * mk[2];
            b2.y  = X[boff[2] + HW] * mk[2];  boff[2] += 4L * HW;
            b3.x  = X[boff[3]]      * mk[3];
            b3.y  = X[boff[3] + HW] * mk[3];  boff[3] += 4L * HW;

            acc0 = __builtin_amdgcn_wmma_f32_16x16x4_f32(
                false, a, false, b0,  (short)0, acc0, false, false);
            acc1 = __builtin_amdgcn_wmma_f32_16x16x4_f32(
                false, a, false, b1v, (short)0, acc1, false, false);
            acc2 = __builtin_amdgcn_wmma_f32_16x16x4_f32(
                false, a, false, b2,  (short)0, acc2, false, false);
            acc3 = __builtin_amdgcn_wmma_f32_16x16x4_f32(
                false, a, false, b3,  (short)0, acc3, false, false);
        }
    }

    // D layout: reg v holds channel m0 + v + 8*hi, pixel n0 + t*16 + ln
    const int chbase = m0 + hi * 8;
#pragma unroll
    for (int t = 0; t < 4; ++t) {
        const v8f acc = (t == 0) ? acc0 : (t == 1) ? acc1 : (t == 2) ? acc2 : acc3;
        long pix = (long)yI[t] * FW_ + xI[t] + ln;
#pragma unroll
        for (int v = 0; v < 8; ++v) {
            int ch = chbase + v;
            Hbuf[((long)(bI[t] * CMID + ch)) * HW + pix] = acc[v] + b1[ch];
        }
    }
}

// ---------------------------------------------------------------------------
// K2: BatchNorm batch statistics (training mode): one block per channel
// ---------------------------------------------------------------------------
__global__ __launch_bounds__(256) void bn_stats_kernel(
    const float* __restrict__ Hbuf, float* __restrict__ mu, float* __restrict__ rstd)
{
    const int c = blockIdx.x;       // 0..127
    float s = 0.0f, s2 = 0.0f;
    for (int i = threadIdx.x; i < NB * HW; i += 256) {
        int b = i / HW, rem = i % HW;
        float v = Hbuf[((long)(b * CMID + c)) * HW + rem];
        s += v;
        s2 += v * v;
    }
    __shared__ float sh[256], sh2[256];
    sh[threadIdx.x] = s;
    sh2[threadIdx.x] = s2;
    __syncthreads();
    for (int off = 128; off > 0; off >>= 1) {
        if (threadIdx.x < off) {
            sh[threadIdx.x]  += sh[threadIdx.x + off];
            sh2[threadIdx.x] += sh2[threadIdx.x + off];
        }
        __syncthreads();
    }
    if (threadIdx.x == 0) {
        const float inv_n = 1.0f / (float)(NB * HW);
        float m = sh[0] * inv_n;
        float var = sh2[0] * inv_n - m * m;
        mu[c] = m;
        rstd[c] = rsqrtf(var + 1e-5f);
    }
}

// ---------------------------------------------------------------------------
// K3: BN-apply + ReLU + Conv2d(128->64, 1x1) via WMMA + softmax
//     + expected-depth + per-group depth partial sums.
//     One wave: all 64 output channels for 16 pixels (4 M-tiles, K=128).
// ---------------------------------------------------------------------------
__global__ __launch_bounds__(32) void head2_kernel(
    const float* __restrict__ Hbuf, const float* __restrict__ mu,
    const float* __restrict__ rstd, const float* __restrict__ gamma,
    const float* __restrict__ beta, const float* __restrict__ W2,
    const float* __restrict__ b2,   const float* __restrict__ bins,
    float* __restrict__ dd, float* __restrict__ ed, float* __restrict__ partial)
{
    const int g = blockIdx.x;       // 16-pixel group, 0..2815
    const int lane = threadIdx.x;
    const int ln = lane & 15, hi = lane >> 4, kk = hi * 2;
    const int n0 = g * 16;
    const int b = n0 / HW;
    const int rem = n0 % HW;
    const int y = rem / FW_;
    const int x0 = rem % FW_;
    const long pix = (long)y * FW_ + x0 + ln;

    v8f acc0 = {}, acc1 = {}, acc2 = {}, acc3 = {};
    const float* hb = Hbuf + ((long)(b * CMID + kk)) * HW + pix;
    const int d0 = ln, d1 = 16 + ln, d2 = 32 + ln, d3 = 48 + ln;

    for (int c0 = 0; c0 < CMID; c0 += 4) {
        const int c = c0 + kk;
        float h0 = hb[0];
        float h1 = hb[HW];
        hb += 4L * HW;
        v2f bf;
        bf.x = fmaxf(0.0f, (h0 - mu[c])     * rstd[c]     * gamma[c]     + beta[c]);
        bf.y = fmaxf(0.0f, (h1 - mu[c + 1]) * rstd[c + 1] * gamma[c + 1] + beta[c + 1]);

        v2f a0, a1, a2, a3;
        a0.x = W2[d0 * CMID + c]; a0.y = W2[d0 * CMID + c + 1];
        a1.x = W2[d1 * CMID + c]; a1.y = W2[d1 * CMID + c + 1];
        a2.x = W2[d2 * CMID + c]; a2.y = W2[d2 * CMID + c + 1];
        a3.x = W2[d3 * CMID + c]; a3.y = W2[d3 * CMID + c + 1];

        acc0 = __builtin_amdgcn_wmma_f32_16x16x4_f32(
            false, a0, false, bf, (short)0, acc0, false, false);
        acc1 = __builtin_amdgcn_wmma_f32_16x16x4_f32(
            false, a1, false, bf, (short)0, acc1, false, false);
        acc2 = __builtin_amdgcn_wmma_f32_16x16x4_f32(
            false, a2, false, bf, (short)0, acc2, false, false);
        acc3 = __builtin_amdgcn_wmma_f32_16x16x4_f32(
            false, a3, false, bf, (short)0, acc3, false, false);
    }

    // logits: reg (t,v) holds channel d = t*16 + v + 8*hi for pixel n0+ln
    float lg[32];
    float mx = -INFINITY;
#pragma unroll
    for (int t = 0; t < 4; ++t) {
        const v8f acc = (t == 0) ? acc0 : (t == 1) ? acc1 : (t == 2) ? acc2 : acc3;
#pragma unroll
        for (int v = 0; v < 8; ++v) {
            const int d = t * 16 + v + hi * 8;
            float val = acc[v] + b2[d];
            lg[t * 8 + v] = val;
            mx = fmaxf(mx, val);
        }
    }
    mx = fmaxf(mx, __shfl_xor(mx, 16, 32));   // combine channel halves

    float s = 0.0f;
#pragma unroll
    for (int i = 0; i < 32; ++i) {
        lg[i] = expf(lg[i] - mx);
        s += lg[i];
    }
    s += __shfl_xor(s, 16, 32);
    const float inv = 1.0f / s;

    float edsum = 0.0f, psum = 0.0f;
#pragma unroll
    for (int t = 0; t < 4; ++t)
#pragma unroll
        for (int v = 0; v < 8; ++v) {
            const int d = t * 16 + v + hi * 8;
            const float p = lg[t * 8 + v] * inv;
            dd[((long)(b * DBINS + d)) * HW + pix] = p;
            edsum += p * bins[d];
            psum  += p;
        }
    edsum += __shfl_xor(edsum, 16, 32);
    if (hi == 0) ed[(long)b * HW + pix] = edsum;

    psum += __shfl_xor(psum, 16, 32);         // full 64-channel sum per pixel
#pragma unroll
    for (int o = 8; o >= 1; o >>= 1)          // sum the 16 pixels of the group
        psum += __shfl_xor(psum, o, 32);
    if (lane == 0) partial[g] = psum;
}

// ---------------------------------------------------------------------------
// K4: deterministic reduce of per-group depth partials -> per-batch depth
//     depth[b] = mean_{H,W}( mean_D softmax ) = sum(partials) / (64*HW)
// ---------------------------------------------------------------------------
__global__ __launch_bounds__(256) void depth_reduce_kernel(
    const float* __restrict__ partial, float* __restrict__ depthB)
{
    const int b = blockIdx.x;       // 0..3
    const int gpb = NGROUPS / NB;   // 704 groups per batch
    float s = 0.0f;
    for (int i = threadIdx.x; i < gpb; i += 256) s += partial[b * gpb + i];
    __shared__ float sh[256];
    sh[threadIdx.x] = s;
    __syncthreads();
    for (int off = 128; off > 0; off >>= 1) {
        if (threadIdx.x < off) sh[threadIdx.x] += sh[threadIdx.x + off];
        __syncthreads();
    }
    if (threadIdx.x == 0)
        depthB[b] = sh[0] / ((float)DBINS * (float)HW);
}

// ---------------------------------------------------------------------------
// K5: lift_to_3d + splat_to_bev (clamp(count,0,1) == non-atomic write of 1.0)
// ---------------------------------------------------------------------------
__global__ __launch_bounds__(256) void lift_splat_kernel(
    const float* __restrict__ uv, const float* __restrict__ Kinv,
    const float* __restrict__ depthB,
    float* __restrict__ p3, float* __restrict__ bev)
{
    const int i = blockIdx.x * blockDim.x + threadIdx.x;
    if (i >= NB * NPTS) return;
    const int b = i / NPTS;
    const int j = i % NPTS;
    const float u = uv[(long)b * (2L * NPTS) + 2L * j + 0];
    const float v = uv[(long)b * (2L * NPTS) + 2L * j + 1];
    const float* Kb = Kinv + b * 9;
    const float d = depthB[b];
    const float px = (u * Kb[0] + v * Kb[1] + Kb[2]) * d;
    const float py = (u * Kb[3] + v * Kb[4] + Kb[5]) * d;
    const float pz = (u * Kb[6] + v * Kb[7] + Kb[8]) * d;
    const long o = (long)i * 3;
    p3[o + 0] = px;
    p3[o + 1] = py;
    p3[o + 2] = pz;
    int gx = (int)((px + 50.0f) * 2.0f);
    int gy = (int)((py + 50.0f) * 2.0f);
    gx = gx < 0 ? 0 : (gx > 199 ? 199 : gx);
    gy = gy < 0 ? 0 : (gy > 199 ? 199 : gy);
    bev[b * 40000 + gy * 200 + gx] = 1.0f;
}

// ---------------------------------------------------------------------------
extern "C" void kernel_launch(void* const* d_in, const int* in_sizes, int n_in,
                              void* d_out, int out_size, void* d_ws, size_t ws_size,
                              hipStream_t stream)
{
    (void)in_sizes; (void)n_in; (void)out_size; (void)ws_size;

    const float* X     = (const float*)d_in[0];  // camera_features
    const float* uv    = (const float*)d_in[1];  // pixels_uv
    const float* Kinv  = (const float*)d_in[2];  // K_inv
    const float* W1    = (const float*)d_in[3];
    const float* b1    = (const float*)d_in[4];
    const float* gamma = (const float*)d_in[5];
    const float* beta  = (const float*)d_in[6];
    const float* W2    = (const float*)d_in[7];
    const float* b2    = (const float*)d_in[8];
    const float* bins  = (const float*)d_in[9];

    // Output layout (concatenated, reference return order)
    float* bev = (float*)d_out;                  // [4,200,200]
    float* dd  = bev + (long)NB * 200 * 200;     // [4,64,64,176]
    float* ed  = dd  + (long)NB * DBINS * HW;    // [4,64,176]
    float* p3  = ed  + (long)NB * HW;            // [4,150000,3]

    // Workspace layout
    float* h       = (float*)d_ws;               // [4,128,64,176] ~23 MB
    float* mu      = h + (long)NB * CMID * HW;   // [128]
    float* rstd    = mu + CMID;                  // [128]
    float* partial = rstd + CMID;                // [2816]
    float* depthB  = partial + NGROUPS;          // [4]

    hipLaunchKernelGGL(zero_bev_kernel, dim3((NB * 200 * 200 + 255) / 256),
                       dim3(256), 0, stream, bev);
    hipLaunchKernelGGL(conv1_wmma_kernel, dim3(NPIX / 64, CMID / 16),
                       dim3(32), 0, stream, X, W1, b1, h);
    hipLaunchKernelGGL(bn_stats_kernel, dim3(CMID), dim3(256), 0, stream,
                       h, mu, rstd);
    hipLaunchKernelGGL(head2_kernel, dim3(NGROUPS), dim3(32), 0, stream,
                       h, mu, rstd, gamma, beta, W2, b2, bins, dd, ed, partial);
    hipLaunchKernelGGL(depth_reduce_kernel, dim3(NB), dim3(256), 0, stream,
                       partial, depthB);
    hipLaunchKernelGGL(lift_splat_kernel, dim3((NB * NPTS + 255) / 256),
                       dim3(256), 0, stream, uv, Kinv, depthB, p3, bev);
}